// OlmoeSparseMoeBlock_25022479466895
// MI455X (gfx1250) — compile-verified
//
#include <hip/hip_runtime.h>
#include <hip/hip_bf16.h>

#define NUM_EXPERTS 64
#define HIDDEN 2048
#define INTER 1024
#define TOP_K 8
#define MAXT 8192
#define BK 128
#define MT 32                      // tokens per block (2 WMMA row-tiles)

typedef __attribute__((ext_vector_type(2))) float v2f;
typedef __attribute__((ext_vector_type(8))) float v8f;

// ---------------- zero outputs + counters ----------------
__global__ void moe_zero_kernel(float* __restrict__ out, size_t n,
                                int* __restrict__ counters, int ncnt) {
    size_t i = (size_t)blockIdx.x * blockDim.x + threadIdx.x;
    size_t stride = (size_t)gridDim.x * blockDim.x;
    for (size_t j = i; j < n; j += stride) out[j] = 0.0f;
    if (i < (size_t)ncnt) counters[i] = 0;
}

// ---------------- router: logits -> softmax -> top8 -> scatter ----------------
__global__ __launch_bounds__(64)
void moe_router_kernel(const float* __restrict__ x,      // [T,H]
                       const float* __restrict__ wr,     // [E,H]
                       int* __restrict__ expert_count,   // [E]
                       int* __restrict__ expert_tokens,  // [E,MAXT]
                       float* __restrict__ expert_prob)  // [E,MAXT]
{
    __shared__ float xrow[HIDDEN];
    __shared__ float p[NUM_EXPERTS];
    __shared__ float q[NUM_EXPERTS];
    __shared__ int   topi[TOP_K];
    __shared__ float topv[TOP_K];

    const int t = blockIdx.x;
    const int e = threadIdx.x;            // blockDim.x == 64
    const float* xr = x + (size_t)t * HIDDEN;
    for (int h = e; h < HIDDEN; h += 64) xrow[h] = xr[h];
    __syncthreads();

    const float* wre = wr + (size_t)e * HIDDEN;
    float acc = 0.0f;
    for (int h = 0; h < HIDDEN; ++h) acc += xrow[h] * wre[h];
    p[e] = acc;
    __syncthreads();

    float m = -INFINITY;
    for (int j = 0; j < NUM_EXPERTS; ++j) m = fmaxf(m, p[j]);
    float ex = __expf(p[e] - m);
    q[e] = ex;
    __syncthreads();
    float s = 0.0f;
    for (int j = 0; j < NUM_EXPERTS; ++j) s += q[j];
    __syncthreads();
    p[e] = ex / s;                        // fp32 softmax probs
    __syncthreads();

    if (e == 0) {
        for (int k = 0; k < TOP_K; ++k) {
            int bi = 0; float bv = -1.0f;
            for (int j = 0; j < NUM_EXPERTS; ++j) {
                float v = p[j];
                if (v > bv) { bv = v; bi = j; }
            }
            topi[k] = bi; topv[k] = bv;
            p[bi] = -1.0f;
        }
    }
    __syncthreads();
    if (e < TOP_K) {
        int sel = topi[e];
        int slot = atomicAdd(&expert_count[sel], 1);
        expert_tokens[(size_t)sel * MAXT + slot] = t;
        expert_prob[(size_t)sel * MAXT + slot] = topv[e];
    }
}

// ---------------- fused expert: gate_up -> silu*up -> down -> scatter-add ----------------
__global__ __launch_bounds__(128)
void moe_expert_kernel(const float* __restrict__ x,        // [T,H]
                       const float* __restrict__ gate_up,  // [E,2I,H]
                       const float* __restrict__ down,     // [E,H,I]
                       const int* __restrict__ expert_count,
                       const int* __restrict__ expert_tokens,
                       const float* __restrict__ expert_prob,
                       float* __restrict__ out)            // [T,H] accumulated
{
    const int e = blockIdx.y;
    const int count = expert_count[e];
    const int m0 = blockIdx.x * MT;
    if (m0 >= count) return;                       // uniform exit

    __shared__ float xs[MT][BK + 4];               // row stride % 64 == 4 banks
    __shared__ float act[MT][INTER + 4];
    __shared__ int   toks[MT];
    __shared__ float probs[MT];

    const int tid = threadIdx.x;
    if (tid < MT) {
        int idx = m0 + tid;
        if (idx < count) {
            toks[tid]  = expert_tokens[(size_t)e * MAXT + idx];
            probs[tid] = expert_prob[(size_t)e * MAXT + idx];
        } else {                                   // pad: reuse a valid row, zero weight
            toks[tid]  = expert_tokens[(size_t)e * MAXT + m0];
            probs[tid] = 0.0f;
        }
    }
    __syncthreads();

    const int wave = tid >> 5;                     // wave32: 4 waves
    const int lane = tid & 31;
    const int mrow = lane & 15;                    // A row (M) and B col (N)
    const int koff = (lane < 16) ? 0 : 2;          // K pair per half-wave (ISA 7.12.2)
    const int rofs = (lane < 16) ? 0 : 8;          // C/D row offset per half-wave

    const float* wgu = gate_up + (size_t)e * (2 * INTER) * HIDDEN;

    // ---- phase 1: act[MT, I] = silu(X Wg^T) * (X Wu^T) ----
    // Each wave owns i-tiles nt = wave + 4*tilei; each B fragment feeds 2 M-tiles.
    for (int tilei = 0; tilei < 16; ++tilei) {
        const int nt = wave + 4 * tilei;           // i-tile 0..63
        const int jg = nt * 16;                    // gate rows
        const int ju = INTER + nt * 16;            // up rows
        v8f accg0 = {}, accg1 = {};
        v8f accu0 = {}, accu1 = {};
        for (int kc = 0; kc < HIDDEN; kc += BK) {
            __syncthreads();
            // cooperative, coalesced X chunk load: MT x BK floats
            for (int idx = tid; idx < MT * (BK / 4); idx += 128) {
                int r  = idx / (BK / 4);
                int c4 = (idx % (BK / 4)) * 4;
                float4 v = *(const float4*)(x + (size_t)toks[r] * HIDDEN + kc + c4);
                *(float4*)&xs[r][c4] = v;
            }
            __syncthreads();
            for (int k = 0; k < BK; k += 4) {
                v2f a0; a0.x = xs[mrow][k + koff];      a0.y = xs[mrow][k + koff + 1];
                v2f a1; a1.x = xs[16 + mrow][k + koff]; a1.y = xs[16 + mrow][k + koff + 1];
                const int gk = kc + k + koff;
                float2 bg = *(const float2*)(wgu + (size_t)(jg + mrow) * HIDDEN + gk);
                v2f b1; b1.x = bg.x; b1.y = bg.y;
                accg0 = __builtin_amdgcn_wmma_f32_16x16x4_f32(
                        false, a0, false, b1, (short)0, accg0, false, false);
                accg1 = __builtin_amdgcn_wmma_f32_16x16x4_f32(
                        false, a1, false, b1, (short)0, accg1, false, false);
                float2 bu = *(const float2*)(wgu + (size_t)(ju + mrow) * HIDDEN + gk);
                v2f b2; b2.x = bu.x; b2.y = bu.y;
                accu0 = __builtin_amdgcn_wmma_f32_16x16x4_f32(
                        false, a0, false, b2, (short)0, accu0, false, false);
                accu1 = __builtin_amdgcn_wmma_f32_16x16x4_f32(
                        false, a1, false, b2, (short)0, accu1, false, false);
            }
        }
#pragma unroll
        for (int r = 0; r < 8; ++r) {
            float g0 = accg0[r], u0 = accu0[r];
            float s0 = g0 / (1.0f + __expf(-g0));
            act[r + rofs][nt * 16 + mrow] = s0 * u0;
            float g1 = accg1[r], u1 = accu1[r];
            float s1 = g1 / (1.0f + __expf(-g1));
            act[16 + r + rofs][nt * 16 + mrow] = s1 * u1;
        }
    }
    __syncthreads();

    // ---- phase 2: out[MT, H] += prob * (act @ Wd^T) ----
    const float* wd = down + (size_t)e * HIDDEN * INTER;
    for (int tilei = 0; tilei < 32; ++tilei) {
        const int nt2 = wave + 4 * tilei;          // h-tile 0..127
        const int h0 = nt2 * 16;
        v8f acc0 = {}, acc1 = {};
        for (int k = 0; k < INTER; k += 4) {
            v2f a0; a0.x = act[mrow][k + koff];      a0.y = act[mrow][k + koff + 1];
            v2f a1; a1.x = act[16 + mrow][k + koff]; a1.y = act[16 + mrow][k + koff + 1];
            float2 bv = *(const float2*)(wd + (size_t)(h0 + mrow) * INTER + k + koff);
            v2f b; b.x = bv.x; b.y = bv.y;
            acc0 = __builtin_amdgcn_wmma_f32_16x16x4_f32(
                    false, a0, false, b, (short)0, acc0, false, false);
            acc1 = __builtin_amdgcn_wmma_f32_16x16x4_f32(
                    false, a1, false, b, (short)0, acc1, false, false);
        }
#pragma unroll
        for (int r = 0; r < 8; ++r) {
            int m = r + rofs;
            float pv0 = probs[m];
            if (pv0 != 0.0f)
                atomicAdd(out + (size_t)toks[m] * HIDDEN + h0 + mrow, pv0 * acc0[r]);
            float pv1 = probs[16 + m];
            if (pv1 != 0.0f)
                atomicAdd(out + (size_t)toks[16 + m] * HIDDEN + h0 + mrow, pv1 * acc1[r]);
        }
    }
}

// ---------------- launch ----------------
extern "C" void kernel_launch(void* const* d_in, const int* in_sizes, int n_in,
                              void* d_out, int out_size, void* d_ws, size_t ws_size,
                              hipStream_t stream) {
    const float* x   = (const float*)d_in[0];   // [B,L,H]
    const float* wr  = (const float*)d_in[1];   // [E,H]
    const float* wgu = (const float*)d_in[2];   // [E,2I,H]
    const float* wd  = (const float*)d_in[3];   // [E,H,I]
    float* out = (float*)d_out;

    const int T = in_sizes[0] / HIDDEN;         // 8192

    char* ws = (char*)d_ws;
    int*   expert_count  = (int*)ws;                                  // 64 ints
    int*   expert_tokens = (int*)(ws + 256);                          // E*MAXT ints
    float* expert_prob   = (float*)(ws + 256 + (size_t)NUM_EXPERTS * MAXT * 4);

    const size_t outN = (size_t)T * HIDDEN;
    moe_zero_kernel<<<1024, 256, 0, stream>>>(out, outN, expert_count, NUM_EXPERTS);
    moe_router_kernel<<<T, 64, 0, stream>>>(x, wr, expert_count, expert_tokens, expert_prob);

    dim3 grid((T + MT - 1) / MT, NUM_EXPERTS);  // x fastest => same-expert tiles adjacent (L2 reuse)
    moe_expert_kernel<<<grid, 128, 0, stream>>>(x, wgu, wd,
                                                expert_count, expert_tokens, expert_prob, out);
}